// CrossTaskConsistencyLoss_43619687858419
// MI455X (gfx1250) — compile-verified
//
#include <hip/hip_runtime.h>
#include <hip/hip_bf16.h>

// ---------------------------------------------------------------------------
// CrossTaskConsistencyLoss for MI455X (gfx1250, wave32).
//
// Strategy:
//   K1 row_scan_wmma : row prefix-sums of seg channels 1,2 via exact-f32
//                      V_WMMA_F32_16X16X4_F32 (D_tile @ upper-triangular ones),
//                      carry chained across 48 column tiles per 16-row stripe.
//   K2 col_scan      : column prefix over the row-prefixed data -> full 2-D
//                      inclusive integral image (L2-resident, coalesced).
//   K3 det_seg       : per-box O(1) SAT queries + block partial sums; also
//                      caches det_conf / det xyxy for the plate kernel.
//   K4 plate         : IoU max over 300 dets per plate + partial sums/counts.
//   K5 final         : single block, fixed-order combine (deterministic).
// ---------------------------------------------------------------------------

typedef __attribute__((ext_vector_type(2))) float v2f;
typedef __attribute__((ext_vector_type(8))) float v8f;

#define B_   16
#define N_   300
#define P_   100
#define L_   32
#define H_   768
#define W_   768
#define NCLS 7

static constexpr int NB_SEG = (B_ * N_ + 255) / 256;   // 19 blocks
static constexpr int NB_PLT = (B_ * P_ + 255) / 256;   // 7 blocks

// workspace layout (float elements)
static constexpr size_t R_ELEMS  = (size_t)B_ * 2 * H_ * W_;      // 18,874,368 (75.5 MB)
static constexpr size_t CONF_OFF = R_ELEMS;                        // det_conf  [B*N]
static constexpr size_t XY_OFF   = CONF_OFF + (size_t)B_ * N_;     // det xyxy  [B*N*4]
static constexpr size_t P0_OFF   = XY_OFF + (size_t)B_ * N_ * 4;   // seg partials
static constexpr size_t P1_OFF   = P0_OFF + 32;                    // plate partials
static constexpr size_t C1_OFF   = P1_OFF + 16;                    // plate counts

// ---------------------------------------------------------------------------
// K1: row-wise inclusive prefix sums using V_WMMA_F32_16X16X4_F32.
// One wave per 16-row stripe of one (b, channel) image; 1536 stripes total.
// P_tile = D_tile(16x16) @ U(16x16), U[k][j] = (k<=j), split into 4 K=4 WMMAs
// accumulated into C, with the inter-tile carry pre-loaded into C.
// ---------------------------------------------------------------------------
__global__ __launch_bounds__(256)
void row_scan_wmma(const float* __restrict__ seg, float* __restrict__ R) {
    const int lane   = threadIdx.x & 31;
    const int wave   = threadIdx.x >> 5;
    const int stripe = blockIdx.x * 8 + wave;        // 0..1535
    const int img    = stripe / (H_ / 16);           // b*2 + c   (c: 0=DRIVEWAY,1=FOOTPATH)
    const int sr     = stripe % (H_ / 16);
    const int b      = img >> 1;
    const int c      = img & 1;
    const int hi     = lane >> 4;                    // lane half (K-group select)
    const int m      = lane & 15;                    // A-matrix row / B,C,D column index

    // source row for A fragments: seg channel c+1, row sr*16 + m
    const size_t srcBase = ((size_t)(b * 3 + (c + 1)) * H_ + (sr * 16 + m)) * (size_t)W_;
    const size_t dstRow0 = ((size_t)img * H_ + sr * 16) * (size_t)W_;

    // Constant B fragments: upper-triangular ones, chunk kc covers K rows
    // kc*4 + 2*hi + {0,1} (mirrors the documented f32 A layout).
    v2f Bk[4];
#pragma unroll
    for (int kc = 0; kc < 4; ++kc) {
        const int k0 = kc * 4 + 2 * hi;
        Bk[kc].x = (k0     <= m) ? 1.0f : 0.0f;
        Bk[kc].y = (k0 + 1 <= m) ? 1.0f : 0.0f;
    }

    float carry[8];
#pragma unroll
    for (int v = 0; v < 8; ++v) carry[v] = 0.0f;

    for (int t = 0; t < W_ / 16; ++t) {
        const int col = t * 16;
        v8f cacc;
#pragma unroll
        for (int v = 0; v < 8; ++v) cacc[v] = carry[v];   // inject running carry

#pragma unroll
        for (int kc = 0; kc < 4; ++kc) {
            const size_t k0 = (size_t)col + kc * 4 + 2 * hi;
            v2f a;
            a.x = seg[srcBase + k0];
            a.y = seg[srcBase + k0 + 1];
            cacc = __builtin_amdgcn_wmma_f32_16x16x4_f32(
                false, a, false, Bk[kc], (short)0, cacc, false, false);
        }

        // C/D layout: VGPR v, lanes 0-15 -> row v, lanes 16-31 -> row v+8; N = lane%16.
#pragma unroll
        for (int v = 0; v < 8; ++v) {
            R[dstRow0 + (size_t)(v + 8 * hi) * W_ + col + m] = cacc[v];
            // new carry for this row = prefix at N=15 (lane 15 of each half)
            carry[v] = __shfl(cacc[v], 15, 16);
        }
    }
}

// ---------------------------------------------------------------------------
// K2: column prefix over row-prefixed data -> full inclusive 2-D integral.
// One thread per column; 32-lane waves touch 32 consecutive columns (coalesced).
// ---------------------------------------------------------------------------
__global__ __launch_bounds__(256)
void col_scan(float* __restrict__ R) {
    const int t = blockIdx.x * blockDim.x + threadIdx.x;
    if (t >= B_ * 2 * W_) return;
    const int img = t / W_;
    const int x   = t % W_;
    const size_t base = (size_t)img * H_ * W_ + x;
    float acc = 0.0f;
    for (int y = 0; y < H_; ++y) {
        const size_t idx = base + (size_t)y * W_;
        acc += R[idx];
        R[idx] = acc;
    }
}

// ---------------------------------------------------------------------------
// K3: per-detection SAT queries + det_conf/xyxy cache + block partial sums.
// ---------------------------------------------------------------------------
__global__ __launch_bounds__(256)
void det_seg_kernel(const float* __restrict__ boxes, const float* __restrict__ logits,
                    float* __restrict__ ws) {
    __shared__ float red[256];
    const float* R = ws;
    const int t = blockIdx.x * 256 + threadIdx.x;
    float term = 0.0f;
    if (t < B_ * N_) {
        const int b = t / N_;
        const float* lg = logits + (size_t)t * NCLS;
        float mx = lg[0];
#pragma unroll
        for (int i = 1; i < NCLS; ++i) mx = fmaxf(mx, lg[i]);
        float s = 0.0f;
#pragma unroll
        for (int i = 0; i < NCLS; ++i) s += expf(lg[i] - mx);
        const float conf = 1.0f / s;                  // max of softmax

        const float* bx = boxes + (size_t)t * 4;
        const float X1 = bx[0] - bx[2] * 0.5f, Y1 = bx[1] - bx[3] * 0.5f;
        const float X2 = bx[0] + bx[2] * 0.5f, Y2 = bx[1] + bx[3] * 0.5f;

        // cache for plate kernel (unscaled xyxy)
        ws[CONF_OFF + t] = conf;
        float* vxy = ws + XY_OFF + (size_t)t * 4;
        vxy[0] = X1; vxy[1] = Y1; vxy[2] = X2; vxy[3] = Y2;

        const int x1 = min(max((int)floorf(X1 * (float)W_), 0), W_ - 1);
        const int y1 = min(max((int)floorf(Y1 * (float)H_), 0), H_ - 1);
        const int x2 = min(max((int)floorf(X2 * (float)W_), 0), W_ - 1);
        const int y2 = min(max((int)floorf(Y2 * (float)H_), 0), H_ - 1);

        if ((conf >= 0.3f) && (x2 > x1) && (y2 > y1)) {
            const float area  = (float)max((y2 - y1) * (x2 - x1), 1);
            const size_t bD = (size_t)(b * 2 + 0) * H_ * W_;   // DRIVEWAY integral
            const size_t bF = (size_t)(b * 2 + 1) * H_ * W_;   // FOOTPATH integral
            auto q = [&](size_t cb, int y, int x) -> float {
                return (y > 0 && x > 0) ? R[cb + (size_t)(y - 1) * W_ + (x - 1)] : 0.0f;
            };
            const float sD = q(bD, y2, x2) - q(bD, y1, x2) - q(bD, y2, x1) + q(bD, y1, x1);
            const float sF = q(bF, y2, x2) - q(bF, y1, x2) - q(bF, y2, x1) + q(bF, y1, x1);
            term = fmaxf(sF / area - sD / area, 0.0f) * conf;
        }
    }
    red[threadIdx.x] = term;
    __syncthreads();
    for (int o = 128; o > 0; o >>= 1) {
        if (threadIdx.x < o) red[threadIdx.x] += red[threadIdx.x + o];
        __syncthreads();
    }
    if (threadIdx.x == 0) ws[P0_OFF + blockIdx.x] = red[0];
}

// ---------------------------------------------------------------------------
// K4: per-plate max-IoU vs valid detections + partial sums / valid counts.
// ---------------------------------------------------------------------------
__global__ __launch_bounds__(256)
void plate_kernel(const float* __restrict__ pboxes, const float* __restrict__ pconfs,
                  float* __restrict__ ws) {
    __shared__ float redS[256];
    __shared__ float redC[256];
    const int t = blockIdx.x * 256 + threadIdx.x;
    float term = 0.0f, cnt = 0.0f;
    if (t < B_ * P_) {
        const int b = t / P_;
        const float pc = pconfs[t];
        if (pc > 0.3f) {                 // p_valid (strict >, per reference)
            cnt = 1.0f;
            const float* pb = pboxes + (size_t)t * 4;
            const float px1 = pb[0] - pb[2] * 0.5f, py1 = pb[1] - pb[3] * 0.5f;
            const float px2 = pb[0] + pb[2] * 0.5f, py2 = pb[1] + pb[3] * 0.5f;
            const float pa = (px2 - px1) * (py2 - py1);
            const float* vconf = ws + CONF_OFF + (size_t)b * N_;
            const float* vxy   = ws + XY_OFF + (size_t)b * N_ * 4;
            float maxiou = -1.0f;
            bool  vany   = false;
            for (int nn = 0; nn < N_; ++nn) {
                const float vc = vconf[nn];
                if (vc > 0.3f) {
                    vany = true;
                    const float vx1 = vxy[nn * 4 + 0], vy1 = vxy[nn * 4 + 1];
                    const float vx2 = vxy[nn * 4 + 2], vy2 = vxy[nn * 4 + 3];
                    const float iw = fmaxf(fminf(px2, vx2) - fmaxf(px1, vx1), 0.0f);
                    const float ih = fmaxf(fminf(py2, vy2) - fmaxf(py1, vy1), 0.0f);
                    const float inter = iw * ih;
                    const float va = (vx2 - vx1) * (vy2 - vy1);
                    maxiou = fmaxf(maxiou, inter / (pa + va - inter + 1e-8f));
                }
            }
            if (vany) term = fmaxf(0.5f - maxiou, 0.0f) * pc;   // b_valid folds into p_valid & vany
        }
    }
    redS[threadIdx.x] = term;
    redC[threadIdx.x] = cnt;
    __syncthreads();
    for (int o = 128; o > 0; o >>= 1) {
        if (threadIdx.x < o) {
            redS[threadIdx.x] += redS[threadIdx.x + o];
            redC[threadIdx.x] += redC[threadIdx.x + o];
        }
        __syncthreads();
    }
    if (threadIdx.x == 0) {
        ws[P1_OFF + blockIdx.x] = redS[0];
        ws[C1_OFF + blockIdx.x] = redC[0];
    }
}

// ---------------------------------------------------------------------------
// K5: deterministic combine (single block; fixed summation order).
// ---------------------------------------------------------------------------
__global__ __launch_bounds__(32)
void final_kernel(const float* __restrict__ pconfs, const float* __restrict__ ocr,
                  const float* __restrict__ ws, float* __restrict__ out) {
    __shared__ float ocrTerm[B_];
    const int tid = threadIdx.x;
    if (tid < B_) {
        float sum = 0.0f;
        for (int l = 0; l < L_; ++l) {
            const float* p = ocr + ((size_t)tid * L_ + l) * 37;
            float m = p[0];
            for (int k = 1; k < 37; ++k) m = fmaxf(m, p[k]);
            sum += m;
        }
        const float avg = sum / (float)L_;
        float mp = pconfs[(size_t)tid * P_];
        for (int k = 1; k < P_; ++k) mp = fmaxf(mp, pconfs[(size_t)tid * P_ + k]);
        ocrTerm[tid] = (avg > 0.7f) ? fmaxf(avg - mp, 0.0f) : 0.0f;
    }
    __syncthreads();
    if (tid == 0) {
        float segs = 0.0f;
        for (int i = 0; i < NB_SEG; ++i) segs += ws[P0_OFF + i];
        float ps = 0.0f, np = 0.0f;
        for (int i = 0; i < NB_PLT; ++i) { ps += ws[P1_OFF + i]; np += ws[C1_OFF + i]; }
        float osum = 0.0f;
        for (int i = 0; i < B_; ++i) osum += ocrTerm[i];
        const float det_seg   = segs / (float)(B_ * N_);
        const float plate_det = ps / fmaxf(np, 1.0f);
        const float ocr_plate = osum / (float)B_;
        out[0] = 0.1f * det_seg + 0.1f * plate_det + 0.1f * ocr_plate;
    }
}

extern "C" void kernel_launch(void* const* d_in, const int* in_sizes, int n_in,
                              void* d_out, int out_size, void* d_ws, size_t ws_size,
                              hipStream_t stream) {
    (void)in_sizes; (void)n_in; (void)out_size; (void)ws_size;
    const float* det_boxes  = (const float*)d_in[0];
    const float* det_logits = (const float*)d_in[1];
    const float* seg        = (const float*)d_in[2];
    const float* pboxes     = (const float*)d_in[3];
    const float* pconf      = (const float*)d_in[4];
    const float* ocr        = (const float*)d_in[5];
    float* out = (float*)d_out;
    float* ws  = (float*)d_ws;   // needs ~75.6 MB (integral of 2 channels + partials)

    // K1: 1536 stripes, 8 waves/block -> 192 blocks
    hipLaunchKernelGGL(row_scan_wmma, dim3(192), dim3(256), 0, stream, seg, ws);
    // K2: 24576 columns
    hipLaunchKernelGGL(col_scan, dim3((B_ * 2 * W_ + 255) / 256), dim3(256), 0, stream, ws);
    // K3: 4800 detections
    hipLaunchKernelGGL(det_seg_kernel, dim3(NB_SEG), dim3(256), 0, stream,
                       det_boxes, det_logits, ws);
    // K4: 1600 plates
    hipLaunchKernelGGL(plate_kernel, dim3(NB_PLT), dim3(256), 0, stream, pboxes, pconf, ws);
    // K5: combine
    hipLaunchKernelGGL(final_kernel, dim3(1), dim3(32), 0, stream, pconf, ocr, ws, out);
}